// BatchCropElements_14173392076849
// MI455X (gfx1250) — compile-verified
//
#include <hip/hip_runtime.h>

#define PROB 0.3f

#define AS1 __attribute__((address_space(1)))
#define AS3 __attribute__((address_space(3)))

typedef int v4i __attribute__((vector_size(16)));  // matches builtin param pointee

#if __has_builtin(__builtin_amdgcn_global_load_async_to_lds_b128) && \
    __has_builtin(__builtin_amdgcn_s_wait_asynccnt)
#define HAVE_GFX1250_ASYNC 1
#else
#define HAVE_GFX1250_ASYNC 0
#endif

// out[b,c,h,w] = x[b,c,h,w] * (rand_u[h,w] > PROB ? 1 : 0)
// x: [64,2048,24,12] f32 (151 MB). Bandwidth-bound: ~302 MB -> ~13 us @ 23.3 TB/s.
// Inner (h,w) period = 288 floats = 72 float4; float4 g uses mask float4 (g % 72).
__global__ __launch_bounds__(256) void batch_mask_mul_f32_kernel(
    const float* __restrict__ x, const float* __restrict__ rand_u,
    float* __restrict__ out, unsigned n4) {
  __shared__ float4 s_rand[72];  // raw rand_u staged in LDS (1152 B)

  const unsigned tid = threadIdx.x;

#if HAVE_GFX1250_ASYNC
  // CDNA5 async copy global->LDS (ASYNCcnt), one b128 per lane for 72 lanes.
  if (tid < 72u) {
    __builtin_amdgcn_global_load_async_to_lds_b128(
        (AS1 v4i*)(rand_u + 4u * tid),
        (AS3 v4i*)(&s_rand[tid]), /*offset=*/0, /*cpol=*/0);
  }
  __builtin_amdgcn_s_wait_asynccnt(0);
#else
  if (tid < 72u) {
    s_rand[tid] = reinterpret_cast<const float4*>(rand_u)[tid];
  }
#endif
  __syncthreads();

  const float4* __restrict__ x4 = reinterpret_cast<const float4*>(x);
  float4* __restrict__ o4 = reinterpret_cast<float4*>(out);

  // Each block owns 1024 contiguous float4s; each thread 4 of them (stride 256)
  // -> 4 independent global_load_b128 in flight per thread, fully coalesced.
  unsigned g = blockIdx.x * 1024u + tid;

  // gfx1250 prefetch path (lowers to global_prefetch_b8).
  __builtin_prefetch(x4 + g, 0, 3);

#pragma unroll
  for (int i = 0; i < 4; ++i, g += 256u) {
    if (g < n4) {
      float4 v = x4[g];
      const float4 r = s_rand[g % 72u];
      // multiply by explicit 1.0/0.0 mask to match reference exactly (incl. -0.0)
      v.x *= (r.x > PROB) ? 1.0f : 0.0f;
      v.y *= (r.y > PROB) ? 1.0f : 0.0f;
      v.z *= (r.z > PROB) ? 1.0f : 0.0f;
      v.w *= (r.w > PROB) ? 1.0f : 0.0f;
      o4[g] = v;
    }
  }
}

extern "C" void kernel_launch(void* const* d_in, const int* in_sizes, int n_in,
                              void* d_out, int out_size, void* d_ws, size_t ws_size,
                              hipStream_t stream) {
  (void)n_in; (void)out_size; (void)d_ws; (void)ws_size;
  const float* x      = (const float*)d_in[0];  // [64,2048,24,12] f32
  const float* rand_u = (const float*)d_in[1];  // [24,12] f32
  float* out = (float*)d_out;

  const unsigned n4 = (unsigned)(in_sizes[0] / 4);     // total float4 count
  const unsigned blocks = (n4 + 1023u) / 1024u;        // 1024 float4 per block

  batch_mask_mul_f32_kernel<<<blocks, 256, 0, stream>>>(x, rand_u, out, n4);
}